// LatentGraph_56324201120420
// MI455X (gfx1250) — compile-verified
//
#include <hip/hip_runtime.h>
#include <hip/hip_bf16.h>

// Sizes fixed by the reference
#define D_IN  128
#define D_LAT 64
#define L_EDG 16
#define E_IN_ 16
#define D_OUT 128

typedef __attribute__((ext_vector_type(2))) float v2f;
typedef __attribute__((ext_vector_type(8))) float v8f;
typedef __attribute__((__vector_size__(4 * sizeof(int)))) int v4i;

static __device__ __forceinline__ v8f wmma_f32(v2f a, v2f b, v8f c) {
  // V_WMMA_F32_16X16X4_F32 : D = A(16x4) * B(4x16) + C(16x16)
  return __builtin_amdgcn_wmma_f32_16x16x4_f32(
      /*neg_a=*/false, a, /*neg_b=*/false, b,
      /*c_mod=*/(short)0, c, /*reuse_a=*/false, /*reuse_b=*/false);
}

// ---- gfx1250 async global->LDS path (guarded so compile can't regress) ----
#if defined(__AMDGCN__) && __has_builtin(__builtin_amdgcn_global_load_async_to_lds_b128)
#define USE_ASYNC_LDS 1
#else
#define USE_ASYNC_LDS 0
#endif

#if USE_ASYNC_LDS
typedef __attribute__((address_space(1))) v4i glb_v4i;
typedef __attribute__((address_space(3))) v4i lds_v4i;
__device__ __forceinline__ void async_ld16(const float* g, float* l) {
  __builtin_amdgcn_global_load_async_to_lds_b128(
      (glb_v4i*)g, (lds_v4i*)l, /*imm_off=*/0, /*cpol=*/0);
}
template <int CNT>
__device__ __forceinline__ void wait_async() {
#if __has_builtin(__builtin_amdgcn_s_wait_asynccnt)
  __builtin_amdgcn_s_wait_asynccnt(CNT);
#else
  if (CNT == 0) asm volatile("s_wait_asynccnt 0x0" ::: "memory");
  else          asm volatile("s_wait_asynccnt 0x1" ::: "memory");
#endif
}
#endif

// ---------------------------------------------------------------- zero scratch
__global__ void k_zero(float4* __restrict__ p, size_t n4) {
  size_t i = (size_t)blockIdx.x * blockDim.x + threadIdx.x;
  size_t stride = (size_t)gridDim.x * blockDim.x;
  float4 z; z.x = z.y = z.z = z.w = 0.f;
  for (; i < n4; i += stride) p[i] = z;
}

// ------------------------------------------- ew = relu(edge_attr @ W_edge + b)
// One wave per 16-edge tile; WMMA f32 16x16x4, K=16.  Epilogue fuses the
// per-channel degree accumulation: deg[col[e]][l] += ew[e][l].
__global__ void k_edge_latent(const float* __restrict__ ea,
                              const float* __restrict__ We,
                              const float* __restrict__ be,
                              const int*   __restrict__ eidx,
                              float* __restrict__ ew,
                              float* __restrict__ deg,
                              int E) {
  int lane  = threadIdx.x & 31;
  int tile  = (int)blockIdx.x * (blockDim.x >> 5) + (threadIdx.x >> 5);
  int ntile = (E + 15) >> 4;
  if (tile >= ntile) return;                 // wave-uniform: EXEC stays full
  int m0   = tile * 16;
  int me   = min(m0 + (lane & 15), E - 1);   // clamped A-row
  int koff = (lane >> 4) << 1;               // 0 or 2
  int nl   = lane & 15;                      // output channel
  v8f acc = {};
#pragma unroll
  for (int kk = 0; kk < E_IN_; kk += 4) {
    v2f a, b;
    a.x = ea[(size_t)me * E_IN_ + kk + koff];
    a.y = ea[(size_t)me * E_IN_ + kk + koff + 1];
    b.x = We[(kk + koff) * L_EDG + nl];
    b.y = We[(kk + koff + 1) * L_EDG + nl];
    acc = wmma_f32(a, b, acc);
  }
  float bias = be[nl];
  int ebase = m0 + ((lane < 16) ? 0 : 8);
  if (m0 + 16 <= E) {                        // wave-uniform fast path: no exec churn
#pragma unroll
    for (int i = 0; i < 8; ++i) {
      int e = ebase + i;
      float v = fmaxf(acc[i] + bias, 0.f);
      ew[(size_t)e * L_EDG + nl] = v;
      int cn = eidx[(size_t)E + e];
      atomicAdd(&deg[(size_t)cn * L_EDG + nl], v);
    }
  } else {
#pragma unroll
    for (int i = 0; i < 8; ++i) {
      int e = ebase + i;
      if (e < E) {
        float v = fmaxf(acc[i] + bias, 0.f);
        ew[(size_t)e * L_EDG + nl] = v;
        int cn = eidx[(size_t)E + e];
        atomicAdd(&deg[(size_t)cn * L_EDG + nl], v);
      }
    }
  }
}

// ------------------------------------------------- dis = rsqrt(deg + 1[self])
__global__ void k_dis(const float* __restrict__ deg, float* __restrict__ dis,
                      size_t n) {
  size_t i = (size_t)blockIdx.x * blockDim.x + threadIdx.x;
  if (i < n) dis[i] = rsqrtf(deg[i] + 1.0f);   // self-loop weight 1, deg+1 >= 1
}

// ------------------------------------------------------------ xW = x @ W_conv
// One wave per 16x16 output tile; K = 128 -> 32 WMMA steps.
__global__ void k_xw(const float* __restrict__ x, const float* __restrict__ Wc,
                     float* __restrict__ xW, int N) {
  int lane  = threadIdx.x & 31;
  int tile  = (int)blockIdx.x * (blockDim.x >> 5) + (threadIdx.x >> 5);
  int mt    = (N + 15) >> 4;
  if (tile >= mt * (D_LAT / 16)) return;     // wave-uniform
  int tm = tile >> 2;                        // D_LAT/16 == 4
  int tn = tile & 3;
  int mr   = min(tm * 16 + (lane & 15), N - 1);
  int koff = (lane >> 4) << 1;
  int n    = tn * 16 + (lane & 15);
  v8f acc = {};
#pragma unroll 4
  for (int k = 0; k < D_IN; k += 4) {
    v2f a, b;
    a.x = x[(size_t)mr * D_IN + k + koff];
    a.y = x[(size_t)mr * D_IN + k + koff + 1];
    b.x = Wc[(k + koff) * D_LAT + n];
    b.y = Wc[(k + koff + 1) * D_LAT + n];
    acc = wmma_f32(a, b, acc);
  }
  int rbase = tm * 16 + ((lane < 16) ? 0 : 8);
  if (tm * 16 + 16 <= N) {                   // wave-uniform fast path
#pragma unroll
    for (int i = 0; i < 8; ++i) xW[(size_t)(rbase + i) * D_LAT + n] = acc[i];
  } else {
#pragma unroll
    for (int i = 0; i < 8; ++i) {
      int row = rbase + i;
      if (row < N) xW[(size_t)row * D_LAT + n] = acc[i];
    }
  }
}

// --------------------------------------------------- edge message scatter-add
// One wave per edge: lanes 0..15 build the 16 channel coefficients
// coeff_l = ew[e,l]*dis[row,l]*dis[col,l]; each lane owns feature cols
// {lane, lane+32}; 16x2 = 32 f32 atomics per lane into cat[col].
__global__ void k_scatter(const int*   __restrict__ eidx,
                          const float* __restrict__ ew,
                          const float* __restrict__ dis,
                          const float* __restrict__ xW,
                          float* __restrict__ cat, int E) {
  int lane = threadIdx.x & 31;
  int wid  = ((int)blockIdx.x * blockDim.x + (int)threadIdx.x) >> 5;
  int nw   = ((int)gridDim.x * blockDim.x) >> 5;
  for (int e = wid; e < E; e += nw) {
    int r = eidx[e];
    int c = eidx[(size_t)E + e];
    float cf = 0.f;
    if (lane < 16)
      cf = ew[(size_t)e * L_EDG + lane] *
           dis[(size_t)r * L_EDG + lane] * dis[(size_t)c * L_EDG + lane];
    float x0 = xW[(size_t)r * D_LAT + lane];
    float x1 = xW[(size_t)r * D_LAT + lane + 32];
    int e2 = e + nw;                          // prefetch next gather row
    if (e2 < E) __builtin_prefetch(&xW[(size_t)eidx[e2] * D_LAT], 0, 1);
    float* dst = cat + (size_t)c * (L_EDG * D_LAT) + lane;
#pragma unroll
    for (int l = 0; l < L_EDG; ++l) {
      float cl = __shfl(cf, l, 32);
      atomicAdd(dst + l * D_LAT,      cl * x0);
      atomicAdd(dst + l * D_LAT + 32, cl * x1);
    }
  }
}

// ------------------- self-loop message + conv bias (per node, no atomics)
__global__ void k_selfloop(const float* __restrict__ dis,
                           const float* __restrict__ xW,
                           const float* __restrict__ bc,
                           float* __restrict__ cat, int N) {
  size_t i = (size_t)blockIdx.x * blockDim.x + threadIdx.x;
  size_t total = (size_t)N * (L_EDG * D_LAT);
  if (i >= total) return;
  int    q = (int)(i & (L_EDG * D_LAT - 1));
  size_t n = i >> 10;                         // /1024
  int l = q >> 6, d = q & 63;
  float ds = dis[n * L_EDG + l];
  cat[i] += ds * ds * xW[n * D_LAT + d] + bc[d];
}

// ------------- out = relu(cat[N,1024] @ W_node[1024,128] + b_node)
// 256-thread block = 8 waves; block owns one 16-row M tile, wave w owns
// columns [16w,16w+16).  A staged through LDS in K-chunks of 64, DOUBLE
// BUFFERED with global_load_async_to_lds_b128 (ASYNCcnt): chunk i+1 is in
// flight while WMMAs consume chunk i.  Each thread issues one B128 async
// load per chunk (16 B, all addresses 16 B-aligned; LDS row stride 68
// floats = 272 B keeps B128 alignment AND conflict-free fragment reads).
__global__ void k_out(const float* __restrict__ cat,
                      const float* __restrict__ Wn,
                      const float* __restrict__ bn,
                      float* __restrict__ out, int N) {
  __shared__ float As[2][16][68];
  const int K = L_EDG * D_LAT;               // 1024
  const int NCHUNK = K / 64;                 // 16
  int lane = threadIdx.x & 31;
  int w    = threadIdx.x >> 5;               // 0..7 -> N tile
  int m0   = (int)blockIdx.x * 16;
  int ml   = lane & 15;
  int koff = (lane >> 4) << 1;
  int n    = w * 16 + ml;
  // cooperative stage assignment: thread t owns row tr, 4 floats at tc
  int tr = threadIdx.x >> 4;                 // 0..15
  int tc = (threadIdx.x & 15) << 2;          // 0,4,...,60
  const float* gsrc = cat + (size_t)min(m0 + tr, N - 1) * K + tc;
  v8f acc = {};
#if USE_ASYNC_LDS
  async_ld16(gsrc, &As[0][tr][tc]);          // prologue: chunk 0 -> buf 0
#endif
  for (int ci = 0; ci < NCHUNK; ++ci) {
    int b = ci & 1;
#if USE_ASYNC_LDS
    if (ci + 1 < NCHUNK) {
      async_ld16(gsrc + (ci + 1) * 64, &As[b ^ 1][tr][tc]);
      wait_async<1>();                       // chunk ci complete; ci+1 in flight
    } else {
      wait_async<0>();
    }
#else
    *(float4*)&As[b][tr][tc] = *(const float4*)(gsrc + ci * 64);
#endif
    __syncthreads();
    int kc = ci * 64;
#pragma unroll
    for (int kk = 0; kk < 64; kk += 4) {
      v2f a, bb;
      a.x = As[b][ml][kk + koff];
      a.y = As[b][ml][kk + koff + 1];
      int kg = kc + kk + koff;
      bb.x = Wn[(size_t)kg * D_OUT + n];
      bb.y = Wn[(size_t)(kg + 1) * D_OUT + n];
      acc = wmma_f32(a, bb, acc);
    }
    __syncthreads();                         // all waves done with buf b
  }
  float bias = bn[n];
#pragma unroll
  for (int i = 0; i < 8; ++i) {
    int row = m0 + ((lane < 16) ? i : (i + 8));
    if (row < N) out[(size_t)row * D_OUT + n] = fmaxf(acc[i] + bias, 0.f);
  }
}

// ---------------------------------------------------------------------- host
extern "C" void kernel_launch(void* const* d_in, const int* in_sizes, int n_in,
                              void* d_out, int out_size, void* d_ws, size_t ws_size,
                              hipStream_t stream) {
  const float* x    = (const float*)d_in[0];
  const int*   eidx = (const int*)  d_in[1];
  const float* ea   = (const float*)d_in[2];
  const float* We   = (const float*)d_in[3];
  const float* be   = (const float*)d_in[4];
  const float* Wc   = (const float*)d_in[5];
  const float* bc   = (const float*)d_in[6];
  const float* Wn   = (const float*)d_in[7];
  const float* bn   = (const float*)d_in[8];
  float* out = (float*)d_out;

  const int N = in_sizes[0] / D_IN;     // 50000
  const int E = in_sizes[1] / 2;        // 1,600,000

  // Workspace layout (floats): [deg N*16][cat N*1024][dis N*16][xW N*64][ew E*16]
  // Total ~= N*1120 + E*16 floats (~326 MB for the reference sizes).
  float* ws  = (float*)d_ws;
  float* deg = ws;
  float* cat = deg + (size_t)N * L_EDG;
  float* dis = cat + (size_t)N * (L_EDG * D_LAT);
  float* xW  = dis + (size_t)N * L_EDG;
  float* ew  = xW  + (size_t)N * D_LAT;

  // 1) zero the accumulators (deg + cat are contiguous)
  size_t zn4 = ((size_t)N * (L_EDG + L_EDG * D_LAT)) / 4;
  k_zero<<<(unsigned)((zn4 + 255) / 256), 256, 0, stream>>>((float4*)deg, zn4);

  // 2) edge latent GEMM (WMMA) + fused degree atomics
  int etiles = (E + 15) >> 4;
  k_edge_latent<<<(unsigned)((etiles + 7) / 8), 256, 0, stream>>>(
      ea, We, be, eidx, ew, deg, E);

  // 3) dis = rsqrt(deg + 1)
  size_t dn = (size_t)N * L_EDG;
  k_dis<<<(unsigned)((dn + 255) / 256), 256, 0, stream>>>(deg, dis, dn);

  // 4) xW = x @ W_conv (WMMA)
  int xtiles = (((N + 15) >> 4)) * (D_LAT / 16);
  k_xw<<<(unsigned)((xtiles + 7) / 8), 256, 0, stream>>>(x, Wc, xW, N);

  // 5) message scatter (dominant phase: ~6.6 GB of f32 atomics)
  k_scatter<<<4096, 256, 0, stream>>>(eidx, ew, dis, xW, cat, E);

  // 6) self-loop term + conv bias
  size_t st = (size_t)N * (L_EDG * D_LAT);
  k_selfloop<<<(unsigned)((st + 255) / 256), 256, 0, stream>>>(dis, xW, bc, cat, N);

  // 7) final GEMM (WMMA, async double-buffered LDS staging)
  k_out<<<(unsigned)((N + 15) / 16), 256, 0, stream>>>(cat, Wn, bn, out, N);
}